// LocalLayer_49271864819851
// MI455X (gfx1250) — compile-verified
//
#include <hip/hip_runtime.h>

typedef __attribute__((ext_vector_type(2))) float v2f;
typedef __attribute__((ext_vector_type(4))) float v4f;
typedef __attribute__((ext_vector_type(8))) float v8f;

#define IN_FEATURES 4096
#define OUT_FEATURES 4096
#define N_WIN 256
#define WIN 64
#define BATCH 8192

#define NW 8              // windows per workgroup (one per wave)
#define BT 32             // batch rows per workgroup (two 16-row WMMA sub-tiles)
#define SPAN 176          // 16*NW + 48 overlap columns staged in LDS
#define LDSTRIDE 180      // pad: 180 floats = 720 B (16B aligned rows, conflict-free b64 reads)

__global__ __launch_bounds__(256)
void local_banded_gemm_wmma(const float* __restrict__ x,
                            const float* __restrict__ W,
                            const float* __restrict__ bias,
                            float* __restrict__ out)
{
    __shared__ float tile[BT * LDSTRIDE];

    const int tid = threadIdx.x;
    const int k0  = blockIdx.x * NW;    // first window of this WG
    const int r0  = blockIdx.y * BT;    // first batch row of this WG

    // Column range covering all 8 clamped windows of this WG
    int wc0 = 16 * k0 - 32;
    wc0 = wc0 < 0 ? 0 : wc0;
    wc0 = wc0 > (IN_FEATURES - SPAN) ? (IN_FEATURES - SPAN) : wc0;

    // ---- cooperative fill: x[r0:r0+32, wc0:wc0+176] -> LDS (float4, fully coalesced) ----
    const int NQ = BT * (SPAN / 4);     // 1408 float4
    for (int i = tid; i < NQ; i += 256) {
        const int row = i / (SPAN / 4);
        const int c4  = i % (SPAN / 4);
        const v4f v = *(const v4f*)(x + (size_t)(r0 + row) * IN_FEATURES + wc0 + c4 * 4);
        *(v4f*)(&tile[row * LDSTRIDE + c4 * 4]) = v;
    }
    __syncthreads();

    const int wave = tid >> 5;
    const int lane = tid & 31;
    const int half = lane >> 4;         // K-half selector per ISA 16x4 f32 A/B layout
    const int ln   = lane & 15;         // M (for A) / N (for B,C,D)

    const int k = k0 + wave;            // this wave's window
    int col0 = 16 * k - 32;             // clamp; W's baked-in mask zeroes the rest
    col0 = col0 < 0 ? 0 : col0;
    col0 = col0 > (IN_FEATURES - WIN) ? (IN_FEATURES - WIN) : col0;
    const int lc0  = col0 - wc0;        // window offset inside the LDS tile
    const int out0 = 16 * k;

    // ---- B tile: W^T[col0:col0+64, out0:out0+16] into 16 x v2f (32 VGPRs) ----
    // B(K=c, N=n) = W[out0+n][col0+c]; lane ln = N, vgpr pair s holds K = 4s+2*half .. +1
    v2f breg[16];
    const float* wrow = W + (size_t)(out0 + ln) * IN_FEATURES + col0 + 2 * half;
    #pragma unroll
    for (int s = 0; s < 16; ++s)
        breg[s] = *(const v2f*)(wrow + 4 * s);

    // ---- accumulators seeded with bias (bias depends only on N = ln) ----
    const float bv = bias[out0 + ln];
    v8f c0, c1;
    #pragma unroll
    for (int i = 0; i < 8; ++i) { c0[i] = bv; c1[i] = bv; }

    // ---- K loop: 16 steps of 16x16x4 f32 WMMA per batch sub-tile ----
    const float* a0p = &tile[ln * LDSTRIDE + lc0 + 2 * half];
    const float* a1p = a0p + 16 * LDSTRIDE;
    #pragma unroll
    for (int s = 0; s < 16; ++s) {
        const v2f a0 = *(const v2f*)(a0p + 4 * s);
        const v2f a1 = *(const v2f*)(a1p + 4 * s);
        c0 = __builtin_amdgcn_wmma_f32_16x16x4_f32(false, a0, false, breg[s],
                                                   (short)0, c0, false, false);
        c1 = __builtin_amdgcn_wmma_f32_16x16x4_f32(false, a1, false, breg[s],
                                                   (short)0, c1, false, false);
    }

    // ---- store: C/D layout -> lane ln = N, vgpr r = M=r (lanes 0-15) / M=r+8 (16-31) ----
    float* o0 = out + (size_t)(r0 + half * 8) * OUT_FEATURES + out0 + ln;
    float* o1 = out + (size_t)(r0 + 16 + half * 8) * OUT_FEATURES + out0 + ln;
    #pragma unroll
    for (int r = 0; r < 8; ++r) {
        o0[(size_t)r * OUT_FEATURES] = c0[r];
        o1[(size_t)r * OUT_FEATURES] = c1[r];
    }
}

extern "C" void kernel_launch(void* const* d_in, const int* in_sizes, int n_in,
                              void* d_out, int out_size, void* d_ws, size_t ws_size,
                              hipStream_t stream) {
    const float* x    = (const float*)d_in[0];
    const float* W    = (const float*)d_in[1];
    const float* bias = (const float*)d_in[2];
    // d_in[3] is the mask -- already baked into W, unused.
    float* out = (float*)d_out;

    dim3 grid(N_WIN / NW, BATCH / BT);   // (32, 256)
    local_banded_gemm_wmma<<<grid, dim3(256), 0, stream>>>(x, W, bias, out);
}